// RelationGCN_26036091748361
// MI455X (gfx1250) — compile-verified
//
#include <hip/hip_runtime.h>
#include <hip/hip_bf16.h>

typedef __attribute__((ext_vector_type(2))) float v2f;
typedef __attribute__((ext_vector_type(8))) float v8f;

#define D 128
#define LN_EPS 1e-5f

// ---------------------------------------------------------------------------
// Kernel 0: L2-normalize relation embeddings: rel = v / max(||v||, 1e-6)
// one wave (32 lanes) per row, lane owns 4 consecutive floats
// ---------------------------------------------------------------------------
__global__ __launch_bounds__(32)
void rgcn_rel_norm(const float* __restrict__ rel_in, float* __restrict__ rel_out, int R) {
    int row  = blockIdx.x;
    int lane = threadIdx.x & 31;
    if (row >= R) return;
    const float4 v = *(const float4*)(rel_in + (size_t)row * D + lane * 4);
    float s = v.x * v.x + v.y * v.y + v.z * v.z + v.w * v.w;
#pragma unroll
    for (int off = 16; off; off >>= 1) s += __shfl_xor(s, off, 32);
    float sc = 1.0f / fmaxf(sqrtf(s), 1e-6f);
    float4 o = {v.x * sc, v.y * sc, v.z * sc, v.w * sc};
    *(float4*)(rel_out + (size_t)row * D + lane * 4) = o;
}

// ---------------------------------------------------------------------------
// Kernel 1: zero a float buffer (grid-stride)
// ---------------------------------------------------------------------------
__global__ __launch_bounds__(256)
void rgcn_zero(float* __restrict__ p, long long n) {
    long long i = (long long)blockIdx.x * blockDim.x + threadIdx.x;
    long long stride = (long long)gridDim.x * blockDim.x;
    for (; i < n; i += stride) p[i] = 0.0f;
}

// ---------------------------------------------------------------------------
// Kernel 2: edge phase.  One wave per edge.
//   h_ref = x[src] - 2*dot(x[src], rel[etype])*rel[etype]
//   acc[tgt] += h_ref ; deg[tgt] += 1
// ---------------------------------------------------------------------------
__global__ __launch_bounds__(256)
void rgcn_edge_scatter(const int* __restrict__ ei,   // [2*E] : row0=src, row1=tgt
                       const int* __restrict__ et,   // [E]
                       const float* __restrict__ xin,
                       const float* __restrict__ rel,
                       float* __restrict__ acc,
                       float* __restrict__ deg,
                       int E) {
    int wave = (int)(((long long)blockIdx.x * blockDim.x + threadIdx.x) >> 5);
    int lane = threadIdx.x & 31;
    if (wave >= E) return;

    int s = ei[wave];
    int t = ei[E + wave];
    int r = et[wave];

    const float4 xs = *(const float4*)(xin + (size_t)s * D + lane * 4);
    const float4 rr = *(const float4*)(rel + (size_t)r * D + lane * 4);

    float pd = xs.x * rr.x + xs.y * rr.y + xs.z * rr.z + xs.w * rr.w;
#pragma unroll
    for (int off = 16; off; off >>= 1) pd += __shfl_xor(pd, off, 32);
    float c = -2.0f * pd;

    float* ap = acc + (size_t)t * D + lane * 4;
    unsafeAtomicAdd(ap + 0, xs.x + c * rr.x);
    unsafeAtomicAdd(ap + 1, xs.y + c * rr.y);
    unsafeAtomicAdd(ap + 2, xs.z + c * rr.z);
    unsafeAtomicAdd(ap + 3, xs.w + c * rr.w);
    if (lane == 0) unsafeAtomicAdd(deg + t, 1.0f);
}

// ---------------------------------------------------------------------------
// Kernel 3: fused  out = LN( (acc/deg) @ W^T + resid ) [relu]
// 4 waves / block, each wave owns a 16-node block and an 8KB LDS slice.
// GEMM via V_WMMA_F32_16X16X4_F32 (exact f32), K-loop 32 x 8 j-tiles.
// ---------------------------------------------------------------------------
__global__ __launch_bounds__(128)
void rgcn_gemm_ln(const float* __restrict__ acc,
                  const float* __restrict__ deg,
                  const float* __restrict__ resid,
                  const float* __restrict__ W,      // [D, D] row-major (out,in)
                  const float* __restrict__ gamma,
                  const float* __restrict__ beta,
                  float* __restrict__ out,
                  int nN, int relu) {
    __shared__ float lds[4][16 * D];                 // 32 KB / block

    const int w    = threadIdx.x >> 5;               // wave in block
    const int lane = threadIdx.x & 31;
    const int kh   = lane >> 4;                      // half-wave selector
    const int nn   = lane & 15;

    const int base = ((int)blockIdx.x * 4 + w) * 16; // first node of this wave
    if (base >= nN) return;                          // wave-uniform: EXEC stays full

    // --- A-fragment row for this lane: m = lane & 15 -------------------------
    int row_a    = base + nn;
    int valid_a  = row_a < nN;
    int row_clmp = valid_a ? row_a : (nN - 1);
    float dg  = deg[row_clmp];
    float idg = valid_a ? (1.0f / fmaxf(dg, 1.0f)) : 0.0f;
    const float* pa = acc + (size_t)row_clmp * D + 2 * kh;   // K offset for this half

    v8f c[8];
#pragma unroll
    for (int jt = 0; jt < 8; ++jt)
#pragma unroll
        for (int v = 0; v < 8; ++v) c[jt][v] = 0.0f;

    // --- K loop: D=128 in steps of 4 ----------------------------------------
    for (int k0 = 0; k0 < D; k0 += 4) {
        v2f a;
        a.x = pa[k0]     * idg;                      // K = k0 + 2*kh + 0
        a.y = pa[k0 + 1] * idg;                      // K = k0 + 2*kh + 1
#pragma unroll
        for (int jt = 0; jt < 8; ++jt) {
            const float* pb = W + (size_t)(jt * 16 + nn) * D + 2 * kh;
            v2f b;
            b.x = pb[k0];                            // B[K, n] = W[n_out, K]
            b.y = pb[k0 + 1];
            c[jt] = __builtin_amdgcn_wmma_f32_16x16x4_f32(
                false, a, false, b, (short)0, c[jt], false, false);
        }
    }

    // --- residual add + stage 16x128 tile to this wave's LDS slice ----------
#pragma unroll
    for (int jt = 0; jt < 8; ++jt) {
#pragma unroll
        for (int v = 0; v < 8; ++v) {
            int row = v + 8 * kh;                    // 0..15
            int col = jt * 16 + nn;                  // 0..127
            int g   = base + row;
            float val = c[jt][v];
            if (g < nN) val += resid[(size_t)g * D + col];
            lds[w][row * D + col] = val;
        }
    }

    // --- LayerNorm per row (all 32 lanes cooperate, 4 cols each) ------------
    const int c0 = lane * 4;
    const float4 g4 = *(const float4*)(gamma + c0);
    const float4 b4 = *(const float4*)(beta + c0);

    for (int row = 0; row < 16; ++row) {             // wave-uniform loop
        if (base + row >= nN) break;                 // wave-uniform guard
        float4 vv = *(float4*)&lds[w][row * D + c0];
        float s = vv.x + vv.y + vv.z + vv.w;
#pragma unroll
        for (int off = 16; off; off >>= 1) s += __shfl_xor(s, off, 32);
        float mu = s * (1.0f / D);
        float d0 = vv.x - mu, d1 = vv.y - mu, d2 = vv.z - mu, d3 = vv.w - mu;
        float q = d0 * d0 + d1 * d1 + d2 * d2 + d3 * d3;
#pragma unroll
        for (int off = 16; off; off >>= 1) q += __shfl_xor(q, off, 32);
        float inv = rsqrtf(q * (1.0f / D) + LN_EPS);
        float4 o;
        o.x = d0 * inv * g4.x + b4.x;
        o.y = d1 * inv * g4.y + b4.y;
        o.z = d2 * inv * g4.z + b4.z;
        o.w = d3 * inv * g4.w + b4.w;
        if (relu) {
            o.x = fmaxf(o.x, 0.0f); o.y = fmaxf(o.y, 0.0f);
            o.z = fmaxf(o.z, 0.0f); o.w = fmaxf(o.w, 0.0f);
        }
        *(float4*)(out + (size_t)(base + row) * D + c0) = o;
    }
}

// ---------------------------------------------------------------------------
extern "C" void kernel_launch(void* const* d_in, const int* in_sizes, int n_in,
                              void* d_out, int out_size, void* d_ws, size_t ws_size,
                              hipStream_t stream) {
    const int*   ei      = (const int*)d_in[0];    // [2, E]
    const int*   et      = (const int*)d_in[1];    // [E]
    const float* x0      = (const float*)d_in[2];  // [N, 128]
    const float* rel_emb = (const float*)d_in[3];  // [R, 128]
    const float* W1      = (const float*)d_in[4];
    const float* W2      = (const float*)d_in[5];
    const float* ln1g    = (const float*)d_in[6];
    const float* ln1b    = (const float*)d_in[7];
    const float* ln2g    = (const float*)d_in[8];
    const float* ln2b    = (const float*)d_in[9];

    const int E = in_sizes[0] / 2;
    const int N = in_sizes[2] / D;
    const int R = in_sizes[3] / D;

    // workspace layout (floats): [rel R*D][deg N][acc N*D][h1 N*D]
    float* rel = (float*)d_ws;
    float* dg  = rel + (size_t)R * D;
    float* acc = dg  + (size_t)N;
    float* h1  = acc + (size_t)N * D;

    const long long zc = (long long)N * D + N;       // acc+deg are contiguous (deg first)
    const int zb = 2048;                             // zero-kernel blocks

    // 0) normalize relation embeddings
    rgcn_rel_norm<<<R, 32, 0, stream>>>(rel_emb, rel, R);

    // ---- layer 1 ----
    rgcn_zero<<<zb, 256, 0, stream>>>(dg, zc);
    {
        long long waves = E;
        int blocks = (int)((waves * 32 + 255) / 256);
        rgcn_edge_scatter<<<blocks, 256, 0, stream>>>(ei, et, x0, rel, acc, dg, E);
    }
    rgcn_gemm_ln<<<(N + 63) / 64, 128, 0, stream>>>(acc, dg, x0, W1, ln1g, ln1b, h1, N, 1);

    // ---- layer 2 ----
    rgcn_zero<<<zb, 256, 0, stream>>>(dg, zc);
    {
        long long waves = E;
        int blocks = (int)((waves * 32 + 255) / 256);
        rgcn_edge_scatter<<<blocks, 256, 0, stream>>>(ei, et, h1, rel, acc, dg, E);
    }
    rgcn_gemm_ln<<<(N + 63) / 64, 128, 0, stream>>>(acc, dg, h1, W2, ln2g, ln2b,
                                                    (float*)d_out, N, 0);
}